// GATv2Model_15496242004817
// MI455X (gfx1250) — compile-verified
//
#include <hip/hip_runtime.h>
#include <math.h>

#define SLOPE 0.2f

typedef float v2f __attribute__((ext_vector_type(2)));
typedef float v8f __attribute__((ext_vector_type(8)));

// ---- monotone float<->uint key for atomic float max via atomicMax(uint) ----
__device__ __forceinline__ unsigned f2key(float f) {
  unsigned u = __float_as_uint(f);
  return (u & 0x80000000u) ? ~u : (u | 0x80000000u);
}
__device__ __forceinline__ float key2f(unsigned k) {
  unsigned u = (k & 0x80000000u) ? (k ^ 0x80000000u) : ~k;
  return __uint_as_float(u);
}
__device__ __forceinline__ float lrelu(float v) { return v > 0.0f ? v : SLOPE * v; }

// ============================================================================
// Y[M,NC] = X[M,K] @ W[K,NC] + b   (M%16==0, NC%16==0)
// one wave per 16x16 tile, V_WMMA_F32_16X16X4_F32 along K.
// A layout: lane l<16 -> row l, K=(k0, k0+1); lanes 16..31 -> K=(k0+2, k0+3).
// B layout: lane l<16 -> col l, rows K=k0 / k0+1 in the two VGPRs; hi half K+2.
// D layout: vgpr v, lane l -> M = v + 8*(l>=16), N = l&15.
// ============================================================================
template <int K, int NC>
__global__ void gemm_bias_wmma(const float* __restrict__ X, const float* __restrict__ W,
                               const float* __restrict__ b, float* __restrict__ Y, int M) {
  const int ntiles = NC / 16;
  int wid = threadIdx.x >> 5;
  int tile = blockIdx.x * (blockDim.x >> 5) + wid;
  int total = (M / 16) * ntiles;
  if (tile >= total) return;                  // wave-uniform: EXEC all-ones for WMMA
  int mt = tile / ntiles, nt = tile % ntiles;
  int lane = threadIdx.x & 31;
  int half = lane >> 4;
  int l = lane & 15;
  int row = mt * 16 + l;
  int col = nt * 16 + l;
  float bc = b[col];
  v8f acc = {bc, bc, bc, bc, bc, bc, bc, bc};
  const float* xp = X + (size_t)row * K;
  const float* wp = W + col;
#pragma unroll
  for (int k0 = 0; k0 < K; k0 += 4) {
    v2f a, bv;
    a.x = xp[k0 + 2 * half];
    a.y = xp[k0 + 2 * half + 1];
    bv.x = wp[(size_t)(k0 + 2 * half) * NC];
    bv.y = wp[(size_t)(k0 + 2 * half + 1) * NC];
    acc = __builtin_amdgcn_wmma_f32_16x16x4_f32(false, a, false, bv, (short)0, acc,
                                                false, false);
  }
  float* yp = Y + (size_t)(mt * 16 + 8 * half) * NC + col;
#pragma unroll
  for (int v = 0; v < 8; v++) yp[(size_t)v * NC] = acc[v];
}

// ============================================================================
// Edge attention logits: e[g] = sum_c lrelu(xl[src][h,c]+xr[dst][h,c])*att[h,c]
// 16 lanes per (edge,head); each lane covers 4 consecutive channels (float4
// gather = coalesced 256B burst per group). Segment-max via atomicMax(uint key).
// ============================================================================
template <int H_, int C_>
__global__ void edge_scores(const float* __restrict__ xl, const float* __restrict__ xr,
                            const float* __restrict__ att, const int* __restrict__ ei,
                            int E_, int Etot, float* __restrict__ ebuf,
                            unsigned* __restrict__ nmax) {
  int g = (int)((blockIdx.x * (size_t)blockDim.x + threadIdx.x) >> 4);
  int sub = threadIdx.x & 15;
  if (g >= Etot * H_) return;
  int eidx = g / H_, head = g % H_;
  int src, dst;
  if (eidx < E_) { src = ei[eidx]; dst = ei[E_ + eidx]; }
  else           { src = dst = eidx - E_; }
  int base = head * C_ + sub * 4;
  float4 a  = *(const float4*)(xl + (size_t)src * (H_ * C_) + base);
  float4 bq = *(const float4*)(xr + (size_t)dst * (H_ * C_) + base);
  float4 at = *(const float4*)(att + base);
  float p = lrelu(a.x + bq.x) * at.x + lrelu(a.y + bq.y) * at.y +
            lrelu(a.z + bq.z) * at.z + lrelu(a.w + bq.w) * at.w;
  // reduce across the 16-lane group (wave32; xor<16 stays inside group)
  for (int off = 8; off > 0; off >>= 1) p += __shfl_xor(p, off, 32);
  if (sub == 0) {
    ebuf[g] = p;
    atomicMax(nmax + (size_t)dst * H_ + head, f2key(p));
  }
}

// ex = exp(e - max[dst]); denom[dst] += ex
template <int H_>
__global__ void edge_exp(const int* __restrict__ ei, int E_, int Etot,
                         const unsigned* __restrict__ nmax, float* __restrict__ ebuf,
                         float* __restrict__ denom) {
  int g = (int)(blockIdx.x * (size_t)blockDim.x + threadIdx.x);
  if (g >= Etot * H_) return;
  int eidx = g / H_, head = g % H_;
  int dst = (eidx < E_) ? ei[E_ + eidx] : (eidx - E_);
  float mx = key2f(nmax[(size_t)dst * H_ + head]);
  float ex = expf(ebuf[g] - mx);
  ebuf[g] = ex;
  atomicAdd(denom + (size_t)dst * H_ + head, ex);
}

// out[dst] += (ex/denom[dst]) * xl[src]  -- 16 lanes x float4 per (edge,head)
template <int H_, int C_>
__global__ void edge_scatter(const float* __restrict__ xl, const int* __restrict__ ei,
                             int E_, int Etot, const float* __restrict__ ebuf,
                             const float* __restrict__ denom, float* __restrict__ oacc) {
  int g = (int)((blockIdx.x * (size_t)blockDim.x + threadIdx.x) >> 4);
  int sub = threadIdx.x & 15;
  if (g >= Etot * H_) return;
  int eidx = g / H_, head = g % H_;
  int src, dst;
  if (eidx < E_) { src = ei[eidx]; dst = ei[E_ + eidx]; }
  else           { src = dst = eidx - E_; }
  float alpha = ebuf[g] / denom[(size_t)dst * H_ + head];
  int base = head * C_ + sub * 4;
  float4 a = *(const float4*)(xl + (size_t)src * (H_ * C_) + base);
  float* o = oacc + (size_t)dst * (H_ * C_) + base;
  atomicAdd(o + 0, alpha * a.x);
  atomicAdd(o + 1, alpha * a.y);
  atomicAdd(o + 2, alpha * a.z);
  atomicAdd(o + 3, alpha * a.w);
}

// in-place: h = elu(h + bias)   (elu alpha=1)
__global__ void bias_elu_inplace(float* __restrict__ h, const float* __restrict__ bias,
                                 int n, int nc) {
  int i = (int)(blockIdx.x * (size_t)blockDim.x + threadIdx.x);
  if (i >= n) return;
  float v = h[i] + bias[i % nc];
  h[i] = v > 0.0f ? v : expm1f(v);
}

__global__ void bias_add_out(const float* __restrict__ acc, const float* __restrict__ bias,
                             float* __restrict__ out, int n, int nc) {
  int i = (int)(blockIdx.x * (size_t)blockDim.x + threadIdx.x);
  if (i >= n) return;
  out[i] = acc[i] + bias[i % nc];
}

extern "C" void kernel_launch(void* const* d_in, const int* in_sizes, int n_in,
                              void* d_out, int out_size, void* d_ws, size_t ws_size,
                              hipStream_t stream) {
  (void)n_in; (void)out_size; (void)ws_size;
  const int IN = 128, HID = 64, H = 4, OUT = 64;
  const float*    x    = (const float*)d_in[0];
  const int*      ei   = (const int*)d_in[1];
  const float*    Wl1  = (const float*)d_in[2];
  const float*    bl1  = (const float*)d_in[3];
  const float*    Wr1  = (const float*)d_in[4];
  const float*    br1  = (const float*)d_in[5];
  const float*    att1 = (const float*)d_in[6];
  const float*    bias1= (const float*)d_in[7];
  const float*    Wl2  = (const float*)d_in[8];
  const float*    bl2  = (const float*)d_in[9];
  const float*    Wr2  = (const float*)d_in[10];
  const float*    br2  = (const float*)d_in[11];
  const float*    att2 = (const float*)d_in[12];
  const float*    bias2= (const float*)d_in[13];
  float* out = (float*)d_out;

  const int N    = in_sizes[0] / IN;   // 50000 (divisible by 16)
  const int E    = in_sizes[1] / 2;    // 400000
  const int Etot = E + N;              // + self loops
  const int HC   = H * HID;            // 256

  // ---- workspace layout (floats) ----
  float* ws = (float*)d_ws;
  size_t o = 0;
  float*    xl1   = ws + o; o += (size_t)N * HC;     // 12.8M
  float*    xr1   = ws + o; o += (size_t)N * HC;     // 12.8M
  float*    hacc  = ws + o; o += (size_t)N * HC;     // 12.8M (becomes h in-place)
  float*    ebuf  = ws + o; o += (size_t)Etot * H;   // 1.8M
  unsigned* nmax  = (unsigned*)(ws + o); o += (size_t)N * H;
  float*    denom = ws + o; o += (size_t)N * H;
  // layer-2 aliases (layer-1 temporaries dead by then)
  float* xl2   = xl1;                       // N*64
  float* xr2   = xr1;                       // N*64
  float* oacc2 = xr1 + (size_t)N * OUT;     // N*64, inside xr1 region

  const int B = 256;

  // ================= layer 1 =================
  hipMemsetAsync(hacc,  0, (size_t)N * HC * sizeof(float), stream);
  hipMemsetAsync(nmax,  0, (size_t)N * H * sizeof(unsigned), stream); // key 0 = -max
  hipMemsetAsync(denom, 0, (size_t)N * H * sizeof(float), stream);

  int tiles1 = (N / 16) * (HC / 16);
  gemm_bias_wmma<128, 256><<<(tiles1 + 7) / 8, B, 0, stream>>>(x, Wl1, bl1, xl1, N);
  gemm_bias_wmma<128, 256><<<(tiles1 + 7) / 8, B, 0, stream>>>(x, Wr1, br1, xr1, N);

  long long items1 = (long long)Etot * H;
  int gb1 = (int)((items1 * 16 + B - 1) / B);
  edge_scores<4, 64><<<gb1, B, 0, stream>>>(xl1, xr1, att1, ei, E, Etot, ebuf, nmax);
  edge_exp<4><<<(int)((items1 + B - 1) / B), B, 0, stream>>>(ei, E, Etot, nmax, ebuf, denom);
  edge_scatter<4, 64><<<gb1, B, 0, stream>>>(xl1, ei, E, Etot, ebuf, denom, hacc);

  int n1 = N * HC;
  bias_elu_inplace<<<(n1 + B - 1) / B, B, 0, stream>>>(hacc, bias1, n1, HC);

  // ================= layer 2 =================
  hipMemsetAsync(oacc2, 0, (size_t)N * OUT * sizeof(float), stream);
  hipMemsetAsync(nmax,  0, (size_t)N * sizeof(unsigned), stream);
  hipMemsetAsync(denom, 0, (size_t)N * sizeof(float), stream);

  int tiles2 = (N / 16) * (OUT / 16);
  gemm_bias_wmma<256, 64><<<(tiles2 + 7) / 8, B, 0, stream>>>(hacc, Wl2, bl2, xl2, N);
  gemm_bias_wmma<256, 64><<<(tiles2 + 7) / 8, B, 0, stream>>>(hacc, Wr2, br2, xr2, N);

  long long items2 = Etot;
  int gb2 = (int)((items2 * 16 + B - 1) / B);
  edge_scores<1, 64><<<gb2, B, 0, stream>>>(xl2, xr2, att2, ei, E, Etot, ebuf, nmax);
  edge_exp<1><<<(int)((items2 + B - 1) / B), B, 0, stream>>>(ei, E, Etot, nmax, ebuf, denom);
  edge_scatter<1, 64><<<gb2, B, 0, stream>>>(xl2, ei, E, Etot, ebuf, denom, oacc2);

  int n2 = N * OUT;
  bias_add_out<<<(n2 + B - 1) / B, B, 0, stream>>>(oacc2, bias2, out, n2, OUT);
}